// SigmaEncoder_86543591014411
// MI455X (gfx1250) — compile-verified
//
#include <hip/hip_runtime.h>
#include <hip/hip_bf16.h>
#include <math.h>

#define B_  512
#define TI_ 1024
#define D_  512
#define LN_EPS 1e-5f

typedef float v2f __attribute__((ext_vector_type(2)));
typedef float v8f __attribute__((ext_vector_type(8)));

// ---------------------------------------------------------------------------
// Kernel 1: fused slot-feature build + exact GELU + masked mean-pool over TI.
// pooledg[b,d] = sum_t mask * gelu(a[b,t]*W1[0,d] + W1[1+mate,d]*act + b1[d])
//               / max(cnt,1)
// ind[b] = (cnt > 0) ? 1 : 0   (controls the b2 bias after the @W2 GEMM)
// One block per batch row b, 512 threads (=16 wave32), thread owns dim d.
// ---------------------------------------------------------------------------
__global__ __launch_bounds__(D_) void pool_gelu_kernel(
    const float* __restrict__ a,     // [B,TI]
    const int*   __restrict__ mate,  // [B,TI]
    const int*   __restrict__ mask,  // [B,TI]
    const float* __restrict__ W1,    // [1+TI, D]
    const float* __restrict__ b1,    // [D]
    float* __restrict__ pooledg,     // [B,D]
    float* __restrict__ ind)         // [B]
{
    __shared__ float a_s[TI_];
    __shared__ int   pk_s[TI_];   // (clipped_mate << 2) | (active << 1) | mask

    const int b = blockIdx.x;
    const int d = threadIdx.x;

    for (int t = d; t < TI_; t += blockDim.x) {
        a_s[t] = a[(size_t)b * TI_ + t];
        int mt  = mate[(size_t)b * TI_ + t];
        int act = (mt >= 0) ? 1 : 0;
        int mc  = (mt < 0) ? 0 : mt;
        int mk  = (mask[(size_t)b * TI_ + t] != 0) ? 1 : 0;
        pk_s[t] = (mc << 2) | (act << 1) | mk;
    }
    __syncthreads();

    const float w1_0 = W1[d];     // row 0 of W1 (multiplies a)
    const float bb1  = b1[d];

    float acc = 0.0f;
    float cnt = 0.0f;
#pragma unroll 4
    for (int t = 0; t < TI_; ++t) {
        const int pk = pk_s[t];
        // prefetch the gather row 8 steps ahead (index is already in LDS)
        const int tf  = (t + 8 < TI_) ? (t + 8) : t;
        const int pkf = pk_s[tf];
        __builtin_prefetch(&W1[(size_t)(1 + (pkf >> 2)) * D_ + d], 0, 3);

        // coalesced gather: all 512 lanes read one contiguous 2KB W1 row
        const float w1g = W1[(size_t)(1 + (pk >> 2)) * D_ + d];
        float h = fmaf(a_s[t], w1_0, bb1);
        h += (pk & 2) ? w1g : 0.0f;                       // * active
        // exact GELU: 0.5*h*(1+erf(h/sqrt(2)))
        const float g = 0.5f * h * (1.0f + erff(h * 0.70710678118654752f));
        const float mf = (float)(pk & 1);
        acc = fmaf(g, mf, acc);
        cnt += mf;
    }
    pooledg[(size_t)b * D_ + d] = acc / fmaxf(cnt, 1.0f);
    if (d == 0) ind[b] = (cnt > 0.0f) ? 1.0f : 0.0f;
}

// ---------------------------------------------------------------------------
// Kernel 2/4: C[512,512] = A[512,512] @ Bm[512,512]
//             + bias[n] * (HAS_ROWSCALE ? rowscale[m] : 1)
// V_WMMA_F32_16X16X4_F32, wave32, full f32 precision.
// Dims hardcoded so A/B offsets fold into immediate global_load offsets.
// Two accumulator chains (k, k+4) double WMMA ILP; summed in epilogue.
// 4 waves per block, one 16x16 output tile per wave.
// ---------------------------------------------------------------------------
template <bool HAS_ROWSCALE>
__global__ __launch_bounds__(128) void wmma_gemm512(
    const float* __restrict__ A,
    const float* __restrict__ Bm,
    const float* __restrict__ bias,
    const float* __restrict__ rowscale,
    float* __restrict__ C)
{
    constexpr int N = 512, K = 512;
    const int wave = threadIdx.x >> 5;
    const int lane = threadIdx.x & 31;
    const int half = lane >> 4;       // 0: lanes 0-15, 1: lanes 16-31
    const int l16  = lane & 15;

    const int tile = blockIdx.x * 4 + wave;   // 32x32 tiles of 16x16
    const int tm = (tile >> 5) << 4;          // output row base
    const int tn = (tile & 31) << 4;          // output col base

    // A row M=tm+l16, starting at this half's K offset (2*half)
    const float* __restrict__ arow = A + (tm + l16) * K + (half << 1);
    // B col N=tn+l16, starting at row K=2*half
    const float* __restrict__ bcol = Bm + (half << 1) * N + tn + l16;

    v8f c0 = {0.f, 0.f, 0.f, 0.f, 0.f, 0.f, 0.f, 0.f};
    v8f c1 = {0.f, 0.f, 0.f, 0.f, 0.f, 0.f, 0.f, 0.f};

#pragma unroll 4
    for (int k = 0; k < K; k += 8) {
        // chain 0: K-slice [k, k+4)
        v2f a0 = *(const v2f*)(arow + k);     // 8B-aligned (even element offset)
        v2f b0;
        b0.x = bcol[k * N];
        b0.y = bcol[k * N + N];
        // chain 1: K-slice [k+4, k+8)
        v2f a1 = *(const v2f*)(arow + k + 4);
        v2f b1;
        b1.x = bcol[(k + 4) * N];
        b1.y = bcol[(k + 4) * N + N];

        c0 = __builtin_amdgcn_wmma_f32_16x16x4_f32(false, a0, false, b0,
                                                   (short)0, c0, false, false);
        c1 = __builtin_amdgcn_wmma_f32_16x16x4_f32(false, a1, false, b1,
                                                   (short)0, c1, false, false);
    }

    const int   n  = tn + l16;
    const float bs = bias[n];
#pragma unroll
    for (int r = 0; r < 8; ++r) {
        const int m = tm + r + (half << 3);   // VGPR r: lo lanes M=r, hi M=r+8
        float v = c0[r] + c1[r];
        if (HAS_ROWSCALE) v += bs * rowscale[m];
        else              v += bs;
        C[(size_t)m * N + n] = v;
    }
}

// ---------------------------------------------------------------------------
// Kernel 3: row-wise LayerNorm over D=512.  One block per row, 512 threads.
// ---------------------------------------------------------------------------
__global__ __launch_bounds__(D_) void layernorm_kernel(
    const float* __restrict__ x,
    const float* __restrict__ g,
    const float* __restrict__ bta,
    float* __restrict__ y)
{
    __shared__ float ws[16], ws2[16];
    const int row = blockIdx.x;
    const int d   = threadIdx.x;
    const float v = x[(size_t)row * D_ + d];

    float s = v, s2 = v * v;
#pragma unroll
    for (int off = 16; off >= 1; off >>= 1) {
        s  += __shfl_down(s,  off, 32);
        s2 += __shfl_down(s2, off, 32);
    }
    const int wv = d >> 5, ln = d & 31;
    if (ln == 0) { ws[wv] = s; ws2[wv] = s2; }
    __syncthreads();
    if (wv == 0) {
        float t  = (ln < 16) ? ws[ln]  : 0.0f;
        float t2 = (ln < 16) ? ws2[ln] : 0.0f;
#pragma unroll
        for (int off = 8; off >= 1; off >>= 1) {
            t  += __shfl_down(t,  off, 32);
            t2 += __shfl_down(t2, off, 32);
        }
        if (ln == 0) { ws[0] = t; ws2[0] = t2; }
    }
    __syncthreads();
    const float mean = ws[0] * (1.0f / D_);
    const float var  = ws2[0] * (1.0f / D_) - mean * mean;
    const float r    = rsqrtf(var + LN_EPS);
    y[(size_t)row * D_ + d] = (v - mean) * r * g[d] + bta[d];
}

// ---------------------------------------------------------------------------
extern "C" void kernel_launch(void* const* d_in, const int* in_sizes, int n_in,
                              void* d_out, int out_size, void* d_ws, size_t ws_size,
                              hipStream_t stream) {
    const float* a    = (const float*)d_in[0];
    const int*   mate = (const int*)  d_in[1];
    const int*   mask = (const int*)  d_in[2];
    const float* W1   = (const float*)d_in[3];
    const float* b1   = (const float*)d_in[4];
    const float* W2   = (const float*)d_in[5];
    const float* b2   = (const float*)d_in[6];
    const float* ln_g = (const float*)d_in[7];
    const float* ln_b = (const float*)d_in[8];
    const float* W3   = (const float*)d_in[9];
    const float* b3   = (const float*)d_in[10];
    float* out = (float*)d_out;

    float* pooledg = (float*)d_ws;            // [B,D]
    float* pooled  = pooledg + B_ * D_;       // [B,D]
    float* normed  = pooled  + B_ * D_;       // [B,D]
    float* ind     = normed  + B_ * D_;       // [B]

    // 1) fused gather + GELU + masked mean-pool  -> pooledg, ind
    pool_gelu_kernel<<<B_, D_, 0, stream>>>(a, mate, mask, W1, b1, pooledg, ind);

    // 2) pooled = pooledg @ W2 + b2 * ind       (WMMA f32 16x16x4)
    const int gemm_blocks = (B_ / 16) * (D_ / 16) / 4;  // 4 waves/block
    wmma_gemm512<true><<<gemm_blocks, 128, 0, stream>>>(pooledg, W2, b2, ind,
                                                        pooled);

    // 3) row LayerNorm -> normed
    layernorm_kernel<<<B_, D_, 0, stream>>>(pooled, ln_g, ln_b, normed);

    // 4) out = normed @ W3 + b3                 (WMMA f32 16x16x4)
    wmma_gemm512<false><<<gemm_blocks, 128, 0, stream>>>(normed, W3, b3, nullptr,
                                                         out);
}